// MSDeformAttnTransformerEncoderLayer_63668595196600
// MI455X (gfx1250) — compile-verified
//
#include <hip/hip_runtime.h>
#include <hip/hip_bf16.h>

typedef __attribute__((ext_vector_type(2))) float v2f;
typedef __attribute__((ext_vector_type(8))) float v8f;
typedef __attribute__((ext_vector_type(4))) unsigned int u32x4;
typedef __attribute__((ext_vector_type(8))) int i32x8;
typedef __attribute__((ext_vector_type(4))) int i32x4;

#define NHEAD 8
#define NLVL  4
#define NPTS  4
#define CDIM  256
#define HDIM  32
#define FDIM  1024
#define LQTOT 21760   // 128*128 + 64*64 + 32*32 + 16*16

// GEMM tiling
#define BM 128
#define BN 64
#define BK 16
#define LDA 20   // A tile LDS row stride (16 data + 4 pad DWORDs) -> TDM pad 3/3
#define LDBN 72  // B tile LDS row stride (64 data + 8 pad DWORDs) -> TDM pad 5/7

// ---------------------------------------------------------------------------
// Issue one 2D TDM tile load (global -> LDS) via the Tensor Data Mover.
// D# packed per CDNA5 ISA ch.8: group0 {count=1, lds_addr, global_addr, type=2},
// group1 {data_size=4B, pad cfg, dims, tile dims, dim0 stride}. Groups 2/3 zero
// (2D tensor; tile_dim2/3 = 0 => unused). Tracked by TENSORcnt.
// This toolchain exposes the 6-arg builtin: (g0, g1, g2, g3, g4, cpol).
// ---------------------------------------------------------------------------
__device__ __forceinline__ void tdm_load_2d(
    unsigned int lds_off_bytes, const float* gptr,
    unsigned int tile0, unsigned int tile1, unsigned int stride0_elems,
    unsigned int pad_interval_code, unsigned int pad_amount_code)
{
    const unsigned long long ga = (unsigned long long)(uintptr_t)gptr;
    u32x4 g0;
    g0[0] = 1u;                                  // count=1, user mode
    g0[1] = lds_off_bytes;                       // LDS byte address
    g0[2] = (unsigned int)(ga & 0xFFFFFFFFull);  // global_addr[31:0]
    g0[3] = (unsigned int)((ga >> 32) & 0x01FFFFFFull) | (2u << 30); // addr[56:32] | type=2

    const unsigned int tdim0 = tile0;            // never OOB: tensor dims == tile dims
    const unsigned int tdim1 = tile1;
    i32x8 g1;
    g1[0] = (int)((2u << 16)                     // data_size = 4 bytes
                | (1u << 20)                     // pad_enable
                | (pad_interval_code << 22)
                | (pad_amount_code << 25));
    g1[1] = (int)((tdim0 & 0xFFFFu) << 16);      // [79:48] tensor_dim0 (lo16)
    g1[2] = (int)(((tdim0 >> 16) & 0xFFFFu) | ((tdim1 & 0xFFFFu) << 16));
    g1[3] = (int)(((tdim1 >> 16) & 0xFFFFu) | ((tile0 & 0xFFFFu) << 16)); // tile_dim0
    g1[4] = (int)(tile1 & 0xFFFFu);              // tile_dim1 (tile_dim2 = 0)
    g1[5] = (int)stride0_elems;                  // tensor_dim0_stride[31:0]
    g1[6] = 0;                                   // stride0[47:32] | stride1[15:0]
    g1[7] = 0;
    const i32x4 z4 = {0, 0, 0, 0};
    const i32x8 z8 = {0, 0, 0, 0, 0, 0, 0, 0};
    __builtin_amdgcn_tensor_load_to_lds(g0, g1, z4, z4, z8, 0);
}

// ---------------------------------------------------------------------------
// Tiled WMMA f32 GEMM:  out[M,N] = act( A[M,K] @ W[K,N] + bias[N] )
// 256 threads = 8 waves; tile BM=128 x BN=64, BK=16; double-buffered LDS fed by
// the Tensor Data Mover (wave 0 issues, TENSORcnt + barrier publishes).
// Each wave owns 32x32 = 2x2 accumulators via V_WMMA_F32_16X16X4_F32.
// ---------------------------------------------------------------------------
template<bool RELU>
__global__ __launch_bounds__(256) void gemm_wmma_f32(
    const float* __restrict__ A, const float* __restrict__ W,
    const float* __restrict__ bias, float* __restrict__ out,
    int M, int N, int K)
{
    __shared__ float lds_a[2][BM * LDA];    // [m][k], TDM-padded rows
    __shared__ float lds_b[2][BK * LDBN];   // [k][n], TDM-padded rows

    const int tid  = threadIdx.x;
    const int lane = tid & 31;
    const int wave = tid >> 5;
    const int wm = (wave & 3) * 32;
    const int wn = (wave >> 2) * 32;
    const int lr = lane & 15;
    const int lh = lane >> 4;
    const size_t m0 = (size_t)blockIdx.x * BM;
    const size_t n0 = (size_t)blockIdx.y * BN;

    v8f acc00 = {}, acc01 = {}, acc10 = {}, acc11 = {};

    const unsigned int a_off[2] = {
        (unsigned int)(uintptr_t)&lds_a[0][0], (unsigned int)(uintptr_t)&lds_a[1][0]};
    const unsigned int b_off[2] = {
        (unsigned int)(uintptr_t)&lds_b[0][0], (unsigned int)(uintptr_t)&lds_b[1][0]};

    const int NT = K / BK;
    if (wave == 0) {
        tdm_load_2d(a_off[0], A + m0 * (size_t)K, BK, BM, (unsigned)K, 3u, 3u);
        tdm_load_2d(b_off[0], W + n0,             BN, BK, (unsigned)N, 5u, 7u);
    }

    for (int t = 0; t < NT; ++t) {
        const int cur = t & 1;
        if (wave == 0) __builtin_amdgcn_s_wait_tensorcnt(0);
        __syncthreads();                       // buf[cur] visible to all waves
        if (wave == 0 && (t + 1) < NT) {
            const int k1 = (t + 1) * BK;
            tdm_load_2d(a_off[cur ^ 1], A + m0 * (size_t)K + k1, BK, BM, (unsigned)K, 3u, 3u);
            tdm_load_2d(b_off[cur ^ 1], W + (size_t)k1 * N + n0, BN, BK, (unsigned)N, 5u, 7u);
        }

        const float* la = lds_a[cur];
        const float* lb = lds_b[cur];
        #pragma unroll
        for (int ks = 0; ks < 4; ++ks) {
            const int koff = ks * 4 + lh * 2;
            // A frag: lane holds (M = lr, K = koff + v)
            v2f a0 = *(const v2f*)(&la[(wm      + lr) * LDA + koff]);
            v2f a1 = *(const v2f*)(&la[(wm + 16 + lr) * LDA + koff]);
            // B frag: lane holds (K = koff + v, N = lr) from [k][n] layout
            v2f b0 = { lb[(koff    ) * LDBN + wn      + lr],
                       lb[(koff + 1) * LDBN + wn      + lr] };
            v2f b1 = { lb[(koff    ) * LDBN + wn + 16 + lr],
                       lb[(koff + 1) * LDBN + wn + 16 + lr] };
            acc00 = __builtin_amdgcn_wmma_f32_16x16x4_f32(false, a0, false, b0, (short)0, acc00, false, false);
            acc01 = __builtin_amdgcn_wmma_f32_16x16x4_f32(false, a0, false, b1, (short)0, acc01, false, false);
            acc10 = __builtin_amdgcn_wmma_f32_16x16x4_f32(false, a1, false, b0, (short)0, acc10, false, false);
            acc11 = __builtin_amdgcn_wmma_f32_16x16x4_f32(false, a1, false, b1, (short)0, acc11, false, false);
        }
        __syncthreads();                       // all reads of buf[cur] done
    }

    // epilogue: D layout => VGPR r : (M = sub_m + lh*8 + r, N = sub_n + lr)
    #pragma unroll
    for (int st = 0; st < 4; ++st) {
        const int sm = (st >> 1) * 16, sn = (st & 1) * 16;
        v8f acc = (st == 0) ? acc00 : (st == 1) ? acc01 : (st == 2) ? acc10 : acc11;
        const size_t om = m0 + wm + sm + lh * 8;
        const size_t on = n0 + wn + sn + lr;
        const float bv = bias[on];
        #pragma unroll
        for (int r = 0; r < 8; ++r) {
            float v = acc[r] + bv;
            if (RELU) v = fmaxf(v, 0.0f);
            out[(om + r) * (size_t)N + on] = v;
        }
    }
}

// ---------------------------------------------------------------------------
// q = src + pos (elementwise, float4)
// ---------------------------------------------------------------------------
__global__ __launch_bounds__(256) void add_vec_kernel(
    const float* __restrict__ a, const float* __restrict__ b,
    float* __restrict__ o, int n4)
{
    const int i = blockIdx.x * blockDim.x + threadIdx.x;
    if (i >= n4) return;
    const float4 va = ((const float4*)a)[i];
    const float4 vb = ((const float4*)b)[i];
    float4 v;
    v.x = va.x + vb.x; v.y = va.y + vb.y; v.z = va.z + vb.z; v.w = va.w + vb.w;
    ((float4*)o)[i] = v;
}

// ---------------------------------------------------------------------------
// Softmax over the 16 (level,point) logits per (row, head). One thread each.
// ---------------------------------------------------------------------------
__global__ __launch_bounds__(256) void softmax16_kernel(float* __restrict__ aw, int rows)
{
    int t = blockIdx.x * blockDim.x + threadIdx.x;
    if (t >= rows) return;
    float* p = aw + (size_t)t * 16;
    float e[16];
    float mx = -3.402823e38f;
    #pragma unroll
    for (int i = 0; i < 16; ++i) mx = fmaxf(mx, p[i]);
    float s = 0.0f;
    #pragma unroll
    for (int i = 0; i < 16; ++i) { e[i] = expf(p[i] - mx); s += e[i]; }
    float inv = 1.0f / s;
    #pragma unroll
    for (int i = 0; i < 16; ++i) p[i] = e[i] * inv;
}

// ---------------------------------------------------------------------------
// Deformable bilinear sampling. One wave per (b, query, head); lane = channel.
// ---------------------------------------------------------------------------
__global__ __launch_bounds__(256) void deform_sample_kernel(
    const float* __restrict__ value, const float* __restrict__ off,
    const float* __restrict__ aw,    const float* __restrict__ refp,
    float* __restrict__ samp, int M, int Lq)
{
    const int wid  = (int)((blockIdx.x * (size_t)blockDim.x + threadIdx.x) >> 5);
    const int lane = threadIdx.x & 31;
    if (wid >= M * NHEAD) return;

    const int h = wid & (NHEAD - 1);
    const int m = wid >> 3;           // b*Lq + q
    const int b = m / Lq;

    const int Hs[NLVL] = {128, 64, 32, 16};
    const int Ws[NLVL] = {128, 64, 32, 16};
    const int ST[NLVL] = {0, 16384, 20480, 21504};

    const float* offp = off + ((size_t)m * NHEAD + h) * (NLVL * NPTS * 2);
    const float* awp  = aw  + ((size_t)m * NHEAD + h) * (NLVL * NPTS);
    const float* refr = refp + (size_t)m * (NLVL * 2);
    const float* vb   = value + ((size_t)b * Lq) * (NHEAD * HDIM) + h * HDIM + lane;

    float acc = 0.0f;
    #pragma unroll
    for (int l = 0; l < NLVL; ++l) {
        const int Hl = Hs[l], Wl = Ws[l];
        const float rx = refr[l * 2 + 0], ry = refr[l * 2 + 1];
        const float* vlev = vb + (size_t)ST[l] * (NHEAD * HDIM);
        #pragma unroll
        for (int p = 0; p < NPTS; ++p) {
            const float ox = offp[(l * NPTS + p) * 2 + 0];
            const float oy = offp[(l * NPTS + p) * 2 + 1];
            const float w  = awp[l * NPTS + p];
            const float px = rx * (float)Wl + ox - 0.5f;   // (rx + ox/Wl)*Wl - 0.5
            const float py = ry * (float)Hl + oy - 0.5f;
            const float x0f = floorf(px), y0f = floorf(py);
            const float lx = px - x0f, ly = py - y0f;
            const int x0 = (int)x0f, y0 = (int)y0f;
            const float wt[4] = {(1.0f - lx) * (1.0f - ly), lx * (1.0f - ly),
                                 (1.0f - lx) * ly,          lx * ly};
            #pragma unroll
            for (int t = 0; t < 4; ++t) {
                const int xi = x0 + (t & 1);
                const int yi = y0 + (t >> 1);
                if (xi >= 0 && xi < Wl && yi >= 0 && yi < Hl) {
                    acc += w * wt[t] * vlev[(size_t)(yi * Wl + xi) * (NHEAD * HDIM)];
                }
            }
        }
    }
    samp[((size_t)m * NHEAD + h) * HDIM + lane] = acc;
}

// ---------------------------------------------------------------------------
// Fused residual + LayerNorm over C=256. One wave per row, 8 elems per lane.
// ---------------------------------------------------------------------------
__global__ __launch_bounds__(256) void add_ln_kernel(
    const float* __restrict__ a, const float* __restrict__ bres,
    const float* __restrict__ g, const float* __restrict__ be,
    float* __restrict__ out, int M)
{
    const int wid  = (int)((blockIdx.x * (size_t)blockDim.x + threadIdx.x) >> 5);
    const int lane = threadIdx.x & 31;
    if (wid >= M) return;

    const float* pa = a    + (size_t)wid * CDIM;
    const float* pb = bres + (size_t)wid * CDIM;
    float x[8];
    float s = 0.0f;
    #pragma unroll
    for (int i = 0; i < 8; ++i) {
        const int c = lane + i * 32;
        x[i] = pa[c] + pb[c];
        s += x[i];
    }
    #pragma unroll
    for (int o = 16; o > 0; o >>= 1) s += __shfl_xor(s, o, 32);
    const float mean = s * (1.0f / 256.0f);
    float v = 0.0f;
    #pragma unroll
    for (int i = 0; i < 8; ++i) { const float d = x[i] - mean; v += d * d; }
    #pragma unroll
    for (int o = 16; o > 0; o >>= 1) v += __shfl_xor(v, o, 32);
    const float inv = rsqrtf(v * (1.0f / 256.0f) + 1e-5f);
    #pragma unroll
    for (int i = 0; i < 8; ++i) {
        const int c = lane + i * 32;
        out[(size_t)wid * CDIM + c] = (x[i] - mean) * inv * g[c] + be[c];
    }
}

// ---------------------------------------------------------------------------
extern "C" void kernel_launch(void* const* d_in, const int* in_sizes, int n_in,
                              void* d_out, int out_size, void* d_ws, size_t ws_size,
                              hipStream_t stream)
{
    const float* src  = (const float*)d_in[0];
    const float* pos  = (const float*)d_in[1];
    const float* refp = (const float*)d_in[2];
    // d_in[3] spatial_shapes, d_in[4] level_start_index: compile-time constants here
    const float* Wv   = (const float*)d_in[5];
    const float* bv   = (const float*)d_in[6];
    const float* Woff = (const float*)d_in[7];
    const float* boff = (const float*)d_in[8];
    const float* Waw  = (const float*)d_in[9];
    const float* baw  = (const float*)d_in[10];
    const float* Wo   = (const float*)d_in[11];
    const float* bo   = (const float*)d_in[12];
    const float* W1   = (const float*)d_in[13];
    const float* b1   = (const float*)d_in[14];
    const float* W2   = (const float*)d_in[15];
    const float* b2   = (const float*)d_in[16];
    const float* g1   = (const float*)d_in[17];
    const float* be1  = (const float*)d_in[18];
    const float* g2   = (const float*)d_in[19];
    const float* be2  = (const float*)d_in[20];

    const int M  = in_sizes[0] / CDIM;   // B * Lq
    const int Lq = LQTOT;

    // workspace layout (floats); MC = M*256, total 6.5*MC
    float* ws = (float*)d_ws;
    const size_t MC = (size_t)M * CDIM;
    float* Q    = ws;                       // [0.0, 1.0) : q = src+pos
    float* V    = ws + MC;                  // [1.0, 2.0) : value
    float* OFF  = ws + 2 * MC;              // [2.0, 3.0) : sampling offsets
    float* AW   = ws + 3 * MC;              // [3.0, 3.5) : attention weights
    float* S    = ws + 3 * MC + MC / 2;     // [3.5, 4.5) : sampled output
    float* ATTN = ws + 4 * MC + MC / 2;     // [4.5, 5.5) : attn proj (later F2)
    float* X    = ws + 5 * MC + MC / 2;     // [5.5, 6.5) : post-LN1
    float* H1   = ws;                       // [0.0, 4.0) : FFN hidden (dead bufs)
    float* F2   = ws + 4 * MC + MC / 2;     // reuse ATTN slot (dead after LN1)

    const dim3 blk(256);

    // 0) q = src + pos
    {
        const int n4 = (int)(MC / 4);
        add_vec_kernel<<<dim3((n4 + 255) / 256), blk, 0, stream>>>(src, pos, Q, n4);
    }
    // 1) value = src @ Wv + bv
    gemm_wmma_f32<false><<<dim3(M / BM, CDIM / BN), blk, 0, stream>>>(
        src, Wv, bv, V, M, CDIM, CDIM);
    // 2) off = q @ Woff + boff
    gemm_wmma_f32<false><<<dim3(M / BM, CDIM / BN), blk, 0, stream>>>(
        Q, Woff, boff, OFF, M, CDIM, CDIM);
    // 3) aw_logits = q @ Waw + baw
    gemm_wmma_f32<false><<<dim3(M / BM, 128 / BN), blk, 0, stream>>>(
        Q, Waw, baw, AW, M, 128, CDIM);
    // 4) softmax over 16 per (row, head)
    {
        const int rows = M * NHEAD;
        softmax16_kernel<<<dim3((rows + 255) / 256), blk, 0, stream>>>(AW, rows);
    }
    // 5) deformable sampling: M*NH waves, 8 waves/block -> M blocks
    deform_sample_kernel<<<dim3(M), blk, 0, stream>>>(V, OFF, AW, refp, S, M, Lq);
    // 6) attn = samp @ Wo + bo
    gemm_wmma_f32<false><<<dim3(M / BM, CDIM / BN), blk, 0, stream>>>(
        S, Wo, bo, ATTN, M, CDIM, CDIM);
    // 7) x = LN(src + attn)
    add_ln_kernel<<<dim3(M / 8), blk, 0, stream>>>(src, ATTN, g1, be1, X, M);
    // 8) h1 = relu(x @ W1 + b1)
    gemm_wmma_f32<true><<<dim3(M / BM, FDIM / BN), blk, 0, stream>>>(
        X, W1, b1, H1, M, FDIM, CDIM);
    // 9) f2 = h1 @ W2 + b2
    gemm_wmma_f32<false><<<dim3(M / BM, CDIM / BN), blk, 0, stream>>>(
        H1, W2, b2, F2, M, CDIM, FDIM);
    // 10) out = LN(x + f2)
    add_ln_kernel<<<dim3(M / 8), blk, 0, stream>>>(X, F2, g2, be2, (float*)d_out, M);
}